// HausdorffDistance_28406913696124
// MI455X (gfx1250) — compile-verified
//
#include <hip/hip_runtime.h>
#include <hip/hip_bf16.h>

// Hausdorff-style loss on a 100x100 grid.
// [S1 | S2] = D @ [1 | py] via V_WMMA_F32_16X16X4_F32, D generated on the fly.
// Column space split CSPLIT ways per 16-row tile for occupancy; per-split
// (S1,S2) pairs land in d_ws, combined by a deterministic reduce kernel.

#define Hh 100
#define Ww 100
#define Nn 10000
#define ROW_TILES 625          // Nn / 16
#define CSPLIT 4               // column-space splits per tile
#define JR_PER 25              // grid rows per split: Hh / CSPLIT
#define KSTEPS 25              // WMMA steps per grid row: Ww / 4
#define TASKS (ROW_TILES * CSPLIT)
#define WAVES_PER_BLOCK 8
#define BLOCK 256

typedef __attribute__((ext_vector_type(2))) float v2f;
typedef __attribute__((ext_vector_type(8))) float v8f;

__global__ __launch_bounds__(BLOCK)
void hausdorff_wmma_kernel(const float* __restrict__ gt,
                           float* __restrict__ ws /* [TASKS][32] */) {
  // py mask staged once per workgroup in LDS (40 KB of the 320 KB WGP pool)
  __shared__ float s_py[Nn];
  for (int idx = threadIdx.x; idx < Nn; idx += BLOCK)
    s_py[idx] = (gt[idx] >= 0.5f) ? 1.0f : 0.0f;
  __syncthreads();

  const int wave = threadIdx.x >> 5;
  const int lane = threadIdx.x & 31;
  const int task = blockIdx.x * WAVES_PER_BLOCK + wave;
  if (task >= TASKS) return;      // wave-uniform: EXEC stays all-ones for WMMA

  const int tile = task >> 2;     // task / CSPLIT
  const int cs   = task & (CSPLIT - 1);

  const int i0 = tile * 16;
  const int m = lane & 15;        // A row within tile == B output-column index
  const int khalf = lane >> 4;    // 0 -> K=0,1   1 -> K=2,3
  const int i = i0 + m;
  const float ri = (float)(i / Ww);
  const float rj = (float)(i % Ww);

  // fixed column offsets of this lane's two A elements within a grid row
  const int offa = khalf * 2;     // K = khalf*2
  const int offb = offa + 1;      // K = khalf*2 + 1

  // dy^2 depends only on (rj, k): precompute once, lives in VGPRs
  float dy2a[KSTEPS], dy2b[KSTEPS];
#pragma unroll
  for (int k = 0; k < KSTEPS; ++k) {
    float da = rj - (float)(offa + 4 * k);
    float db = rj - (float)(offb + 4 * k);
    dy2a[k] = da * da;
    dy2b[k] = db * db;
  }

  // loop-invariant B lane masks: output col 0 = ones, col 1 = py
  const float b0 = (m == 0) ? 1.0f : 0.0f;
  const float b1 = (m == 1) ? 1.0f : 0.0f;

  v8f acc = {};
  const int jr0 = cs * JR_PER;
  float dx = ri - (float)jr0;
  const float* pyrow = &s_py[jr0 * Ww];

#pragma unroll 1
  for (int jr = 0; jr < JR_PER; ++jr) {
    const float dx2 = dx * dx;
#pragma unroll
    for (int k = 0; k < KSTEPS; ++k) {
      v2f a, b;
      a.x = __builtin_amdgcn_sqrtf(dx2 + dy2a[k]);   // raw v_sqrt_f32
      a.y = __builtin_amdgcn_sqrtf(dx2 + dy2b[k]);
      const float pya = pyrow[offa + 4 * k];          // ds_load offset:imm
      const float pyb = pyrow[offb + 4 * k];
      b.x = b0 + b1 * pya;                            // one v_fma each
      b.y = b0 + b1 * pyb;
      acc = __builtin_amdgcn_wmma_f32_16x16x4_f32(
          /*neg_a=*/false, a, /*neg_b=*/false, b,
          /*c_mod=*/(short)0, acc, /*reuse_a=*/false, /*reuse_b=*/false);
    }
    dx -= 1.0f;
    pyrow += Ww;
  }

  // C layout: VGPR r holds (M=r, N=lane) in lanes 0-15 and (M=8+r, N=lane-16)
  // in lanes 16-31.  S1 at N=0 (lanes 0,16), S2 at N=1 (lanes 1,17).
  if (m < 2) {                       // lanes 0,1,16,17 write
    const int col = m;               // 0 -> S1, 1 -> S2
    const int rbase = khalf * 8;
    float* wsl = ws + (size_t)task * 32;
#pragma unroll
    for (int r = 0; r < 8; ++r)
      wsl[(rbase + r) * 2 + col] = acc[r];
  }
}

// Deterministic reduction: combine CSPLIT partial (S1,S2) per row, apply px,
// fixed-order accumulate, scale by 1/N^2 (mean over j and over i).
__global__ __launch_bounds__(BLOCK)
void hausdorff_reduce_kernel(const float* __restrict__ prob,
                             const float* __restrict__ ws,
                             float* __restrict__ out) {
  __shared__ float s_part[BLOCK];
  const int t = threadIdx.x;
  float local = 0.0f;
  for (int i = t; i < Nn; i += BLOCK) {
    const int tile = i >> 4;
    const int r = i & 15;
    const float* base = ws + (size_t)tile * CSPLIT * 32 + r * 2;
    float S1 = 0.0f, S2 = 0.0f;
#pragma unroll
    for (int cs = 0; cs < CSPLIT; ++cs) {
      S1 += base[cs * 32 + 0];
      S2 += base[cs * 32 + 1];
    }
    const float px = (prob[i] >= 0.5f) ? 1.0f : 0.0f;
    local += fabsf(px * S1 - S2);
  }
  s_part[t] = local;
  __syncthreads();
  if (t == 0) {
    float s = 0.0f;
#pragma unroll 1
    for (int k = 0; k < BLOCK; ++k) s += s_part[k];
    out[0] = s * (1.0f / ((float)Nn * (float)Nn));
  }
}

extern "C" void kernel_launch(void* const* d_in, const int* in_sizes, int n_in,
                              void* d_out, int out_size, void* d_ws, size_t ws_size,
                              hipStream_t stream) {
  const float* prob = (const float*)d_in[0];  // prob_map [1,100,100] f32
  const float* gt   = (const float*)d_in[1];  // gt_map   [1,100,100] f32
  float* out = (float*)d_out;                 // scalar f32
  float* ws  = (float*)d_ws;                  // TASKS*32 f32 = 320 KB scratch

  const int grid = (TASKS + WAVES_PER_BLOCK - 1) / WAVES_PER_BLOCK; // 313
  hausdorff_wmma_kernel<<<grid, BLOCK, 0, stream>>>(gt, ws);
  hausdorff_reduce_kernel<<<1, BLOCK, 0, stream>>>(prob, ws, out);
}